// ResNet_37426345018001
// MI455X (gfx1250) — compile-verified
//
#include <hip/hip_runtime.h>
#include <hip/hip_bf16.h>
#include <math.h>

// Problem constants (match reference)
#define NE   5            // experts
#define CIN  256
#define COUT 256
#define KS   3
#define HID  65           // CIN/4 + 1
#define BB   32           // batch
#define HH   56
#define WW   56
#define HW   (HH * WW)        // 3136
#define KDIM (CIN * KS * KS)  // 2304 (K-dim of implicit GEMM; also weight cols)
#define WSZ  (COUT * KDIM)    // 589824 floats per expert / per sample

typedef __attribute__((ext_vector_type(2))) float v2f;
typedef __attribute__((ext_vector_type(8))) float v8f;

// ---------------------------------------------------------------------------
// 1) Global average pool: s[b*CIN + c] = mean(x[b,c,:,:])  (float4 loads)
// ---------------------------------------------------------------------------
__global__ void pool_kernel(const float* __restrict__ x, float* __restrict__ s) {
    __shared__ float red[256];
    const int bc = blockIdx.x;                  // b*CIN + c
    const float4* p = (const float4*)(x + (size_t)bc * HW);
    float sum = 0.f;
    for (int i = threadIdx.x; i < HW / 4; i += 256) {   // 784 float4's
        float4 v = p[i];
        sum += v.x + v.y + v.z + v.w;
    }
    red[threadIdx.x] = sum;
    __syncthreads();
    for (int off = 128; off > 0; off >>= 1) {
        if ((int)threadIdx.x < off) red[threadIdx.x] += red[threadIdx.x + off];
        __syncthreads();
    }
    if (threadIdx.x == 0) s[bc] = red[0] * (1.0f / HW);
}

// ---------------------------------------------------------------------------
// 2) Attention MLP: h = relu(s @ w1^T); attn = sigmoid(h @ w2^T + b2)
// ---------------------------------------------------------------------------
__global__ void attn_kernel(const float* __restrict__ s,
                            const float* __restrict__ w1,
                            const float* __restrict__ w2,
                            const float* __restrict__ b2,
                            float* __restrict__ attn) {
    __shared__ float hsh[BB * HID];
    for (int idx = threadIdx.x; idx < BB * HID; idx += blockDim.x) {
        const int b = idx / HID, j = idx % HID;
        const float* sb = s + b * CIN;
        const float* wj = w1 + j * CIN;
        float acc = 0.f;
        for (int c = 0; c < CIN; ++c) acc += sb[c] * wj[c];
        hsh[idx] = fmaxf(acc, 0.f);
    }
    __syncthreads();
    for (int idx = threadIdx.x; idx < BB * NE; idx += blockDim.x) {
        const int b = idx / NE, e = idx % NE;
        const float* hb = hsh + b * HID;
        const float* we = w2 + e * HID;
        float acc = b2[e];
        for (int j = 0; j < HID; ++j) acc += hb[j] * we[j];
        attn[idx] = 1.0f / (1.0f + __expf(-acc));
    }
}

// ---------------------------------------------------------------------------
// 3) Per-expert weight alignment GEMM with WMMA f32 16x16x4:
//    wal[e] (COUT x KDIM) = amat[e] (COUT x COUT) @ wexp[e] (COUT x KDIM)
// ---------------------------------------------------------------------------
__global__ void align_kernel(const float* __restrict__ amat,
                             const float* __restrict__ wexp,
                             float* __restrict__ wal) {
    const int NT = KDIM / 16;   // 144
    const int MT = COUT / 16;   // 16
    const int wave = blockIdx.x * (blockDim.x >> 5) + (threadIdx.x >> 5);
    const int lane = threadIdx.x & 31;
    const int e   = wave / (MT * NT);
    const int rem = wave % (MT * NT);
    const int mt  = rem / NT, nt = rem % NT;

    const float* A  = amat + (size_t)e * COUT * COUT;  // [j][o]
    const float* Bm = wexp + (size_t)e * COUT * KDIM;  // [o][n]

    const int m    = lane & 15;
    const int koff = (lane >> 4) << 1;     // lanes 16-31 hold K+2, K+3
    const int col  = nt * 16 + (lane & 15);
    const float* Arow = A + (size_t)(mt * 16 + m) * COUT;

    v8f acc = {};
    for (int k0 = 0; k0 < COUT; k0 += 4) {
        const int k = k0 + koff;
        v2f a, b;
        a.x = Arow[k];
        a.y = Arow[k + 1];
        b.x = Bm[(size_t)k * KDIM + col];
        b.y = Bm[(size_t)(k + 1) * KDIM + col];
        acc = __builtin_amdgcn_wmma_f32_16x16x4_f32(
            false, a, false, b, (short)0, acc, false, false);
    }
    float* D = wal + (size_t)e * COUT * KDIM;
    const int rbase = mt * 16 + ((lane >> 4) << 3);   // M = r + 8 for lanes 16-31
    #pragma unroll
    for (int r = 0; r < 8; ++r)
        D[(size_t)(rbase + r) * KDIM + col] = acc[r];
}

// ---------------------------------------------------------------------------
// 4) Per-sample weight aggregation: aggw[b][n] = sum_e attn[b][e]*wal[e][n]
//    float4-vectorized (b128 loads/stores).
// ---------------------------------------------------------------------------
__global__ void agg_kernel(const float* __restrict__ attn,
                           const float* __restrict__ wal,
                           float* __restrict__ aggw) {
    __shared__ float at[BB * NE];
    for (int i = threadIdx.x; i < BB * NE; i += blockDim.x) at[i] = attn[i];
    __syncthreads();
    const int n4 = blockIdx.x * blockDim.x + threadIdx.x;   // float4 index
    if (n4 >= WSZ / 4) return;
    const float4* w4 = (const float4*)wal;
    float4 we[NE];
    #pragma unroll
    for (int e = 0; e < NE; ++e) we[e] = w4[(size_t)e * (WSZ / 4) + n4];
    float4* o4 = (float4*)aggw;
    for (int b = 0; b < BB; ++b) {
        const float* ab = at + b * NE;
        float4 v;
        v.x = ab[0]*we[0].x + ab[1]*we[1].x + ab[2]*we[2].x + ab[3]*we[3].x + ab[4]*we[4].x;
        v.y = ab[0]*we[0].y + ab[1]*we[1].y + ab[2]*we[2].y + ab[3]*we[3].y + ab[4]*we[4].y;
        v.z = ab[0]*we[0].z + ab[1]*we[1].z + ab[2]*we[2].z + ab[3]*we[3].z + ab[4]*we[4].z;
        v.w = ab[0]*we[0].w + ab[1]*we[1].w + ab[2]*we[2].w + ab[3]*we[3].w + ab[4]*we[4].w;
        o4[(size_t)b * (WSZ / 4) + n4] = v;
    }
}

// ---------------------------------------------------------------------------
// CDNA5 async memory->LDS copy (ASYNCcnt-tracked), per CDNA5 ISA.
// vdst VGPR = LDS byte address (LDS aperture keeps offset in addr[31:0]).
// ---------------------------------------------------------------------------
__device__ __forceinline__ void async_load_f32_to_lds(const float* gaddr,
                                                      unsigned lds_addr) {
    asm volatile("global_load_async_to_lds_b32 %0, %1, off"
                 :: "v"(lds_addr), "v"(gaddr) : "memory");
}
__device__ __forceinline__ void wait_asynccnt0() {
    asm volatile("s_wait_asynccnt 0x0" ::: "memory");
}

// ---------------------------------------------------------------------------
// 5) Per-sample 3x3 conv as implicit GEMM with WMMA f32 16x16x4.
//    One block = (b, 112-pixel tile); 16 waves = 16 COUT tiles; each wave
//    owns 7 n-tiles (7 accumulators) so every A fragment (global_load_b64)
//    feeds 7 WMMAs.  The im2col B-panel is double-buffered in LDS and filled
//    with EXEC-predicated global_load_async_to_lds_b32 (no VGPR round-trip);
//    padding slots are chunk-invariant and pre-zeroed once.  B fragments are
//    conflict-free ds_load_b64 from the pair-interleaved layout.
// ---------------------------------------------------------------------------
#define JT     7                    // n-tiles per wave
#define NB     (16 * JT)            // 112 pixels per block
#define NBLK   (HW / NB)            // 28 tiles per sample
#define CB     8                    // input channels per LDS chunk
#define KC     (CB * KS * KS)       // 72 k-values per chunk
#define FILL_N (KC * NB)            // 8064 LDS floats per chunk
#define FILL_A ((FILL_N + 511) / 512) // 16 fill slots per thread

__global__ __launch_bounds__(512)
void conv_kernel(const float* __restrict__ x,
                 const float* __restrict__ aggw,
                 float* __restrict__ y) {
    // pair-interleaved: element (k, p) at ((k>>1)*2*NB + p*2 + (k&1))
    __shared__ float bsh[2][FILL_N];

    const int tid  = threadIdx.x;
    const int wave = tid >> 5;         // 0..15 == mt (COUT tile)
    const int lane = tid & 31;
    const int b   = blockIdx.x / NBLK;
    const int ntb = blockIdx.x % NBLK;

    const int nl   = lane & 15;
    const int koff = (lane >> 4) << 1;   // lanes 16-31 hold K+2, K+3

    // ---- precompute fill source offsets (invariant across channel chunks) --
    int xoff[FILL_A];
    #pragma unroll
    for (int a = 0; a < FILL_A; ++a) {
        const int lidx = tid + a * 512;    // LDS slot this thread fills
        int off = -1;
        if (lidx < FILL_N) {
            const int bit0 = lidx & 1;
            const int t    = lidx >> 1;
            const int kp   = t / NB;
            const int p    = t - kp * NB;
            const int k    = kp * 2 + bit0;     // 0..KC-1
            const int cik  = k / 9;             // channel within chunk
            const int r9   = k - 9 * cik;
            const int ky   = r9 / 3;
            const int kx   = r9 - 3 * ky;
            const int pp   = ntb * NB + p;      // global pixel
            const int ph   = pp / WW, pw = pp - ph * WW;
            const int ih   = ph + ky - 1;
            const int iw   = pw + kx - 1;
            if ((unsigned)ih < (unsigned)HH && (unsigned)iw < (unsigned)WW)
                off = cik * HW + ih * WW + iw;
        }
        xoff[a] = off;
    }

    // ---- pre-zero the (chunk-invariant) padding slots in both buffers -----
    #pragma unroll
    for (int a = 0; a < FILL_A; ++a) {
        const int lidx = tid + a * 512;
        if (lidx < FILL_N && xoff[a] < 0) {
            bsh[0][lidx] = 0.f;
            bsh[1][lidx] = 0.f;
        }
    }

    const float* Aw = aggw + (size_t)b * WSZ + (size_t)(wave * 16 + nl) * KDIM;
    const float* xb = x + (size_t)b * CIN * HW;

    v8f acc[JT] = {};

    // ---- prologue: async-fill chunk 0 into buffer 0 ------------------------
    {
        const float* xc = xb;   // cb = 0
        #pragma unroll
        for (int a = 0; a < FILL_A; ++a) {
            const int lidx = tid + a * 512;
            if (lidx < FILL_N && xoff[a] >= 0)
                async_load_f32_to_lds(xc + xoff[a],
                                      (unsigned)(uintptr_t)&bsh[0][lidx]);
        }
    }
    wait_asynccnt0();
    __syncthreads();

    int buf = 0;
    for (int cb = 0; cb < CIN; cb += CB, buf ^= 1) {
        // ---- prefetch next chunk into the other buffer (async, no wait) ---
        if (cb + CB < CIN) {
            const float* xc = xb + (size_t)(cb + CB) * HW;
            #pragma unroll
            for (int a = 0; a < FILL_A; ++a) {
                const int lidx = tid + a * 512;
                if (lidx < FILL_N && xoff[a] >= 0)
                    async_load_f32_to_lds(xc + xoff[a],
                                          (unsigned)(uintptr_t)&bsh[buf ^ 1][lidx]);
            }
        }

        // ---- compute on current buffer ------------------------------------
        const float* bufp = bsh[buf];
        const float* Ak = Aw + cb * 9;
        #pragma unroll 2
        for (int k0 = 0; k0 < KC; k0 += 4) {   // 18 k-steps/chunk
            const int k = k0 + koff;           // even
            v2f a;
            a.x = Ak[k];
            a.y = Ak[k + 1];
            const float* bp = &bufp[(k >> 1) * (2 * NB) + (nl << 1)];
            #pragma unroll
            for (int j = 0; j < JT; ++j) {     // A fragment reused 7x
                v2f bb;
                bb.x = bp[j * 32];
                bb.y = bp[j * 32 + 1];
                acc[j] = __builtin_amdgcn_wmma_f32_16x16x4_f32(
                    false, a, false, bb, (short)0, acc[j], false, false);
            }
        }

        wait_asynccnt0();      // my prefetch for next chunk has landed
        __syncthreads();       // everyone done reading buf & filling buf^1
    }

    float* yb = y + (size_t)b * COUT * HW;
    const int rbase = wave * 16 + ((lane >> 4) << 3);  // M = r+8 for lanes 16-31
    #pragma unroll
    for (int j = 0; j < JT; ++j) {
        const int col = ntb * NB + j * 16 + nl;
        #pragma unroll
        for (int r = 0; r < 8; ++r)
            yb[(size_t)(rbase + r) * HW + col] = acc[j][r];
    }
}

// ---------------------------------------------------------------------------
extern "C" void kernel_launch(void* const* d_in, const int* in_sizes, int n_in,
                              void* d_out, int out_size, void* d_ws, size_t ws_size,
                              hipStream_t stream) {
    (void)in_sizes; (void)n_in; (void)out_size; (void)ws_size;
    const float* x    = (const float*)d_in[0];  // (B, CIN, H, W)
    const float* wexp = (const float*)d_in[1];  // (E, COUT, CIN, 3, 3)
    const float* amat = (const float*)d_in[2];  // (E, COUT, COUT)
    const float* w1   = (const float*)d_in[3];  // (HID, CIN)
    const float* w2   = (const float*)d_in[4];  // (E, HID)
    const float* b2   = (const float*)d_in[5];  // (E,)
    float* y  = (float*)d_out;                  // (B, COUT, H, W)
    float* ws = (float*)d_ws;

    float* s    = ws;                              // 8192 floats
    float* attn = ws + 8192;                       // 160 floats
    float* wal  = ws + 8448;                       // E*WSZ   = 2,949,120 floats
    float* aggw = wal + (size_t)NE * WSZ;          // B*WSZ   = 18,874,368 floats

    hipLaunchKernelGGL(pool_kernel, dim3(BB * CIN), dim3(256), 0, stream, x, s);
    hipLaunchKernelGGL(attn_kernel, dim3(1), dim3(256), 0, stream, s, w1, w2, b2, attn);

    // align: waves = E * (COUT/16) * (KDIM/16) = 5*16*144 = 11520; 8 waves/block
    hipLaunchKernelGGL(align_kernel, dim3(11520 / 8), dim3(256), 0, stream,
                       amat, wexp, wal);

    hipLaunchKernelGGL(agg_kernel, dim3((WSZ / 4 + 255) / 256), dim3(256), 0, stream,
                       attn, wal, aggw);

    // conv: one block per (b, 112-px tile): 32 * 28 = 896 blocks x 512 thr
    hipLaunchKernelGGL(conv_kernel, dim3(BB * NBLK), dim3(512), 0, stream,
                       x, aggw, y);
}